// ITASoftmax_3564822856120
// MI455X (gfx1250) — compile-verified
//
#include <hip/hip_runtime.h>
#include <stdint.h>

// ITA integer streaming softmax, x:(2,16,2048,2048) int32 in [-128,127] -> uint8 (stored as float).
// One wave32 per row of K=2048. Row staged global->LDS with CDNA5 async loads (ASYNCcnt),
// GVS (saddr) addressing + immediate offsets, non-temporal hints on both streams.
//
// Per row (WIDTH=16, 128 chunks):
//   gmax_j = prefix-max of chunk maxes (wave prefix scan)
//   S_j    = sum_i 256 >> (gmax_j - x_i)                 (parallel, per-lane)
//   e_j    = (e_{j-1} >> (gmax_j - gmax_{j-1})) + S_j    (exact 128-step fold, floor semantics)
//   out    = floor(65280 / e_final) >> (gmax_final - x)

#define ROW_LEN 2048
#define WAVES_PER_BLOCK 8
#define CHUNKS 128  // ROW_LEN / 16

typedef float vfloat4 __attribute__((ext_vector_type(4)));

__device__ __forceinline__ int imax(int a, int b) { return a > b ? a : b; }

// 256 >> s with s>=32 -> 0 ; s in [0,255] (256>>9..31 == 0 anyway)
__device__ __forceinline__ int expterm(int s) { return (s < 32) ? (256 >> s) : 0; }

__global__ __launch_bounds__(256) void ita_softmax_kernel(const int* __restrict__ x,
                                                          float* __restrict__ out,
                                                          int rows) {
  __shared__ int lds_row[WAVES_PER_BLOCK * ROW_LEN];  // 64 KB row staging
  __shared__ int fold_buf[WAVES_PER_BLOCK * CHUNKS];  // 4 KB (d<<16)|S per chunk

  const int wave = threadIdx.x >> 5;
  const int lane = threadIdx.x & 31;
  const int row  = blockIdx.x * WAVES_PER_BLOCK + wave;
  const bool live = (row < rows);  // wave-uniform

  // ---------------- async global -> LDS stage of this wave's 8KB row ----------------
  if (live) {
    // generic pointer to __shared__: low 32 bits are the LDS byte offset (ISA 10.2 aperture)
    unsigned ldst = (unsigned)(size_t)(const void*)(&lds_row[wave * ROW_LEN]) + (unsigned)(lane * 16);
    unsigned voff = (unsigned)(lane * 16);
    // wave-uniform 64-bit row base -> SGPR pair for the GVS (saddr) form
    unsigned long long gb = (unsigned long long)(const void*)(x + (size_t)row * ROW_LEN);
    unsigned glo = __builtin_amdgcn_readfirstlane((unsigned)gb);
    unsigned ghi = __builtin_amdgcn_readfirstlane((unsigned)(gb >> 32));
    unsigned long long gs = ((unsigned long long)ghi << 32) | glo;
    // INST_OFFSET is added to BOTH the LDS (VDST) and global (SADDR+VADDR) addresses.
#pragma unroll
    for (int it = 0; it < 16; ++it) {
      asm volatile("global_load_async_to_lds_b128 %0, %1, %2 offset:%3 th:TH_LOAD_NT"
                   :: "v"(ldst), "v"(voff), "s"(gs), "i"(it * 512)
                   : "memory");
    }
    asm volatile("s_wait_asynccnt 0x0" ::: "memory");
  }

  int M = -128;  // final row max
  int invq = 0;  // floor(65280 / eps_sum)

  if (live) {
    // Lane l owns chunks 4l..4l+3 == elements [64l, 64l+64). Keep them in registers.
    const int4* vp = (const int4*)(&lds_row[wave * ROW_LEN + lane * 64]);
    int4 v[16];
#pragma unroll
    for (int i = 0; i < 16; ++i) v[i] = vp[i];

    // per-chunk maxes
    int cm[4];
#pragma unroll
    for (int k = 0; k < 4; ++k) {
      int m = -2147483647;
#pragma unroll
      for (int q = 0; q < 4; ++q) {
        int4 a = v[4 * k + q];
        m = imax(m, imax(imax(a.x, a.y), imax(a.z, a.w)));
      }
      cm[k] = m;
    }
    // in-lane prefix max over the 4 owned chunks
    int pm[4];
    pm[0] = cm[0];
    pm[1] = imax(pm[0], cm[1]);
    pm[2] = imax(pm[1], cm[2]);
    pm[3] = imax(pm[2], cm[3]);

    // wave-level inclusive prefix max of per-lane totals (Hillis-Steele, wave32)
    int incl = pm[3];
#pragma unroll
    for (int d = 1; d < 32; d <<= 1) {
      int t = __shfl_up(incl, d, 32);
      if (lane >= d) incl = imax(incl, t);
    }
    int excl = __shfl_up(incl, 1, 32);
    if (lane == 0) excl = -128;  // initial gmax

    int mk[4];
    mk[0] = imax(excl, pm[0]);
    mk[1] = imax(excl, pm[1]);
    mk[2] = imax(excl, pm[2]);
    mk[3] = imax(excl, pm[3]);
    M = __shfl(incl, 31, 32);  // global row max

    // per-chunk partial sums against the prefix max, pack (d<<16)|S for the fold
#pragma unroll
    for (int k = 0; k < 4; ++k) {
      int s = 0;
#pragma unroll
      for (int q = 0; q < 4; ++q) {
        int4 a = v[4 * k + q];
        s += expterm(mk[k] - a.x);
        s += expterm(mk[k] - a.y);
        s += expterm(mk[k] - a.z);
        s += expterm(mk[k] - a.w);
      }
      int prev = (k == 0) ? excl : mk[k - 1];
      int dd = mk[k] - prev;  // >= 0, <= 255
      fold_buf[wave * CHUNKS + 4 * lane + k] = (dd << 16) | s;  // S <= 4096 fits 16 bits
    }
  }

  __syncthreads();  // fold_buf visible to all lanes of each wave

  if (live) {
    // exact sequential fold (floor shifts are order-dependent); redundantly on all
    // lanes -- every lane reads the same LDS address (broadcast, conflict-free).
    int e = 0;
#pragma unroll 4
    for (int j = 0; j < CHUNKS; ++j) {
      int w = fold_buf[wave * CHUNKS + j];
      int dd = w >> 16;
      int s  = w & 0xFFFF;
      e = ((dd < 32) ? (e >> dd) : 0) + s;
    }
    // eps_sum >= 256 always (global-max element contributes 256 with no later rescale)
    invq = 65280 / e;  // floor((2^8-1)*2^8 / eps_sum), <= 255

    // ---------------- pass 2: coalesced non-temporal output ----------------
    float* orow = out + (size_t)row * ROW_LEN;
#pragma unroll
    for (int it = 0; it < 16; ++it) {
      int4 a = *(const int4*)(&lds_row[wave * ROW_LEN + it * 128 + lane * 4]);
      vfloat4 r;
      {
        int sh = M - a.x;
        r.x = (float)((sh < 32) ? (invq >> sh) : 0);
      }
      {
        int sh = M - a.y;
        r.y = (float)((sh < 32) ? (invq >> sh) : 0);
      }
      {
        int sh = M - a.z;
        r.z = (float)((sh < 32) ? (invq >> sh) : 0);
      }
      {
        int sh = M - a.w;
        r.w = (float)((sh < 32) ? (invq >> sh) : 0);
      }
      // 512B per wave store, fully coalesced; NT: write-once stream, keep it out of L2
      __builtin_nontemporal_store(r, (vfloat4*)(&orow[it * 128 + lane * 4]));
    }
  }
}

extern "C" void kernel_launch(void* const* d_in, const int* in_sizes, int n_in,
                              void* d_out, int out_size, void* d_ws, size_t ws_size,
                              hipStream_t stream) {
  const int* x = (const int*)d_in[0];
  float* out = (float*)d_out;
  int rows = in_sizes[0] / ROW_LEN;  // 65536 for (2,16,2048,2048)
  dim3 block(256);
  dim3 grid((rows + WAVES_PER_BLOCK - 1) / WAVES_PER_BLOCK);
  ita_softmax_kernel<<<grid, block, 0, stream>>>(x, out, rows);
}